// PDFNormal_46179488367043
// MI455X (gfx1250) — compile-verified
//
#include <hip/hip_runtime.h>

typedef __attribute__((ext_vector_type(2))) float v2f;
typedef __attribute__((ext_vector_type(8))) float v8f;
typedef __attribute__((ext_vector_type(4))) int v4i;

#define AS1 __attribute__((address_space(1)))
#define AS3 __attribute__((address_space(3)))

#define EPSV 1e-8f
#define LOG2E 1.4426950408889634f
#define TILE_M 16          // rows per wave-tile
#define DCOLS 128          // feature dimension
#define ROW_PAD 132        // LDS row stride in floats (bank-conflict-free, 16B aligned)
#define WAVES_PER_BLOCK 4

#if __has_builtin(__builtin_amdgcn_global_load_async_to_lds_b128)
#define HAVE_ASYNC_LDS 1
#endif

#if __has_builtin(__builtin_amdgcn_exp2f)
#define EXP2F(v) __builtin_amdgcn_exp2f(v)   // raw v_exp_f32 (2^x)
#else
#define EXP2F(v) exp2f(v)
#endif

__global__ __launch_bounds__(128) void pdfnormal_wmma_kernel(
    const float* __restrict__ x,
    const float* __restrict__ mean,
    const float* __restrict__ var,
    float* __restrict__ out) {
  __shared__ float xs[WAVES_PER_BLOCK][TILE_M * ROW_PAD];
  // A[j] = sqrt(0.5*log2e/(var+eps)); Bp[j] = -A[j]*m[j]
  // exp(-0.5*(x-m)^2/ve) == exp2( -(fma(A,x,Bp))^2 )
  __shared__ float tA[DCOLS];
  __shared__ float tBp[DCOLS];

  const int lane = threadIdx.x & 31;
  const int wave = threadIdx.x >> 5;
  const int tile = blockIdx.x * WAVES_PER_BLOCK + wave;
  const long long row0 = (long long)tile * TILE_M;

  // ---- kick off async staging of this wave's 16x128 tile into LDS ----
#ifdef HAVE_ASYNC_LDS
  {
    AS1 v4i* gsrc = (AS1 v4i*)(x + row0 * DCOLS) + lane;
    AS3 v4i* ldst = (AS3 v4i*)(&xs[wave][0]) + lane;
#pragma unroll
    for (int i = 0; i < TILE_M; ++i) {
      __builtin_amdgcn_global_load_async_to_lds_b128(
          gsrc + i * (DCOLS / 4), ldst + i * (ROW_PAD / 4), 0, 0);
    }
  }
#endif

  // ---- per-wave constants: var+eps, total sum (wave32 butterfly) ----
  float4 vv = *(const float4*)(var + 4 * lane);
  float4 mm = *(const float4*)(mean + 4 * lane);
  float ve0 = vv.x + EPSV, ve1 = vv.y + EPSV, ve2 = vv.z + EPSV, ve3 = vv.w + EPSV;
  float lsum = (ve0 + ve1) + (ve2 + ve3);
#pragma unroll
  for (int m = 1; m < 32; m <<= 1) lsum += __shfl_xor(lsum, m, 32);
  const float inv_total = 1.0f / lsum;

  if (wave == 0) {
    float a0 = __builtin_sqrtf((0.5f * LOG2E) / ve0);
    float a1 = __builtin_sqrtf((0.5f * LOG2E) / ve1);
    float a2 = __builtin_sqrtf((0.5f * LOG2E) / ve2);
    float a3 = __builtin_sqrtf((0.5f * LOG2E) / ve3);
    *(float4*)(&tA[4 * lane]) = make_float4(a0, a1, a2, a3);
    *(float4*)(&tBp[4 * lane]) =
        make_float4(-a0 * mm.x, -a1 * mm.y, -a2 * mm.z, -a3 * mm.w);
  }

#ifdef HAVE_ASYNC_LDS
#if __has_builtin(__builtin_amdgcn_s_wait_asynccnt)
  __builtin_amdgcn_s_wait_asynccnt(0);
#else
  asm volatile("s_wait_asynccnt 0x0" ::: "memory");
#endif
#else
  // fallback: manual coalesced b128 staging through VGPRs
  {
    const float4* xg = (const float4*)(x + row0 * DCOLS);
#pragma unroll
    for (int i = 0; i < TILE_M; ++i) {
      float4 v = xg[i * (DCOLS / 4) + lane];
      *(float4*)(&xs[wave][i * ROW_PAD + 4 * lane]) = v;
    }
  }
#endif

  __syncthreads();  // tables visible to all waves; all LDS staging complete

  // ---- compute probs in WMMA A-layout; per-chunk WMMA does the reduction
  // (accumulation rides the matrix pipe, off the VALU issue stream).
  // A layout (16x4 f32): lane l -> row (l&15); VGPR0/1 -> K = 2*(l>>4) + {0,1}
  const int h = lane >> 4;
  const int r = lane & 15;
  const float* xrow = &xs[wave][r * ROW_PAD];
  v8f acc = {};
  v2f bscale;
  bscale[0] = inv_total;   // fold 1/sum(var+eps) into B: D = inv_total * rowsum
  bscale[1] = inv_total;
#pragma unroll
  for (int c = 0; c < DCOLS / 4; ++c) {
    const int c0 = 4 * c + 2 * h;
    float2 xv = *(const float2*)(xrow + c0);   // ds_load_b64, conflict-free
    float2 av = *(const float2*)(tA + c0);     // 2addr-fusable (512B apart),
    float2 bv = *(const float2*)(tBp + c0);    //   broadcast within half-wave
    float u0 = __builtin_fmaf(av.x, xv.x, bv.x);
    float u1 = __builtin_fmaf(av.y, xv.y, bv.y);
    float p0 = EXP2F(-(u0 * u0));              // neg folds into v_exp_f32 src
    float p1 = EXP2F(-(u1 * u1));
    v2f a;
    a[0] = p0;
    a[1] = p1;
    acc = __builtin_amdgcn_wmma_f32_16x16x4_f32(false, a, false, bscale,
                                                (short)0, acc, false, false);
  }

  // ---- extract: VGPR k holds row k (lanes 0-15) / row k+8 (lanes 16-31),
  // replicated across all 16 columns. Dynamic-select acc[lane&7], then one
  // bpermute pulls rows 8..15 from the upper half into lanes 8..15.
  const int k7 = lane & 7;
  float sel = acc[0];
#pragma unroll
  for (int k = 1; k < 8; ++k)
    if (k7 == k) sel = acc[k];
  const int srcl = (lane & 8) ? (lane + 16) : lane;  // only lanes 0-15 matter
  float val = __shfl(sel, srcl & 31, 32);            // single ds_bpermute

  if (lane < 16) out[row0 + lane] = val;
}

extern "C" void kernel_launch(void* const* d_in, const int* in_sizes, int n_in,
                              void* d_out, int out_size, void* d_ws, size_t ws_size,
                              hipStream_t stream) {
  const float* x = (const float*)d_in[0];
  const float* mean = (const float*)d_in[1];
  const float* var = (const float*)d_in[2];
  float* out = (float*)d_out;

  const int n_rows = out_size;                              // N = 1048576
  const int blocks = n_rows / (TILE_M * WAVES_PER_BLOCK);   // 16384
  pdfnormal_wmma_kernel<<<blocks, 128, 0, stream>>>(x, mean, var, out);
}